// SqRL_64458869178979
// MI455X (gfx1250) — compile-verified
//
#include <hip/hip_runtime.h>
#include <cstdint>

// ---------------------------------------------------------------------------
// Square rotational unrolling gather, computed-index version.
//   in : x (B=16, C=32, H=512, W=512) f32 ; out: (B, C, 256, 2048) f32
// Pure data movement (~1.5 GiB => ~64 us floor at 23.3 TB/s).
// One block per (image, ring-row). The ring's two hot rows are staged into
// LDS by the gfx1250 Tensor Data Mover (TENSOR_LOAD_TO_LDS + s_wait_tensorcnt);
// the unrolled ring reads them back with explicit ds_load_b32 (inline asm --
// the TDM's LDS write is invisible to LLVM alias analysis, which otherwise
// deletes compiler-generated LDS loads no matter how the array is "defined").
// Column segments stream from global (image is L2-resident); output uses
// NT b128 stores (write-once, keep L2 for the input).
// ---------------------------------------------------------------------------

typedef float        v4f   __attribute__((ext_vector_type(4)));
typedef unsigned int u32x4 __attribute__((ext_vector_type(4)));
typedef int          i32x4 __attribute__((ext_vector_type(4)));
typedef int          i32x8 __attribute__((ext_vector_type(8)));

#define HH    512          // H == W
#define HR    256          // ceil(H/2)
#define RL    2048         // 4*H   (output row length)
#define LMID  255          // (H-1)/2
#define WRAP  2044         // 4*H-4 (tail wrap point)

#if defined(__gfx1250__) && __has_builtin(__builtin_amdgcn_tensor_load_to_lds)
#define USE_TDM 1
#else
#define USE_TDM 0
#endif

// Explicit LDS read: ds_load_b32 from a byte offset, drained before use.
// Guaranteed to emit on the DS pipe regardless of what alias analysis thinks.
__device__ __forceinline__ float lds_read_f32(uint32_t byte_off)
{
    float v;
    asm volatile("ds_load_b32 %0, %1\n\t"
                 "s_wait_dscnt 0x0"
                 : "=v"(v) : "v"(byte_off) : "memory");
    return v;
}

__global__ __launch_bounds__(512)
void sqrl_gather_kernel(const float* __restrict__ x, float* __restrict__ out)
{
    const int blk = blockIdx.x;            // img*HR + dif
    const int dif = blk & (HR - 1);
    const int img = blk >> 8;
    const int i   = LMID - dif;            // ring coordinate (from border)
    const int el  = 2 * dif + 1;           // edge run length (H even)
    const int bot = i + el;                // == H-1-i

    const float* __restrict__ src = x + (size_t)img * (HH * HH);
    float* __restrict__ dst = out + (size_t)blk * RL;

    __shared__ float rows[2][HH];          // [0]=row i, [1]=row H-1-i

    const int tid = threadIdx.x;
    const uint32_t ldsb = (uint32_t)(uintptr_t)(&rows[0][0]);  // LDS byte base

#if USE_TDM
    if (tid < 32) {                        // wave 0 issues one TDM descriptor
        const uint64_t gaddr = (uint64_t)(uintptr_t)(src + (size_t)i * HH);

        // ---- D# group 0 (128b): count | lds_addr | global_addr | type=2 ----
        u32x4 g0;
        g0[0] = 1u;                                          // count=1, user D#
        g0[1] = ldsb;                                        // lds_addr (bytes)
        g0[2] = (uint32_t)(gaddr & 0xFFFFFFFFu);             // global_addr[31:0]
        g0[3] = (uint32_t)((gaddr >> 32) & 0x1FFFFFFu)       // global_addr[56:32]
              | (2u << 30);                                  // type = 2 ("image")

        // ---- D# group 1 (256b): 2-D tile, 512 x 2 lines, stride el*H ----
        i32x8 g1;
        g1[0] = (2 << 16);           // workgroup_mask=0, data_size=2 (4 bytes)
        g1[1] = (HH << 16);          // tensor_dim0[15:0] = 512
        g1[2] = (2 << 16);           // tensor_dim0[31:16]=0 ; tensor_dim1 = 2
        g1[3] = (HH << 16);          // tensor_dim1[31:16]=0 ; tile_dim0 = 512
        g1[4] = 2;                   // tile_dim1 = 2 ; tile_dim2 = 0
        g1[5] = el * HH;             // tensor_dim0_stride[31:0]  (elements)
        g1[6] = 0;                   // stride[47:32]=0 ; dim1_stride lo = 0
        g1[7] = 0;

        i32x4 gz4 = {0, 0, 0, 0};                    // groups 2/3 unused (<=2-D)
        i32x8 gz8 = {0, 0, 0, 0, 0, 0, 0, 0};        // trailing group (clang-23 form)

        // 6-arg amdgpu-toolchain signature:
        // (uint32x4, int32x8, int32x4, int32x4, int32x8, i32 cpol)
        __builtin_amdgcn_tensor_load_to_lds(g0, g1, gz4, gz4, gz8, /*cpol=*/0);

#if __has_builtin(__builtin_amdgcn_s_wait_tensorcnt)
        __builtin_amdgcn_s_wait_tensorcnt(0);
#else
        asm volatile("s_wait_tensorcnt 0x0" ::: "memory");
#endif
        asm volatile("" ::: "memory");
    }

    // Never-taken at runtime (opaque predicate); exists only as a real IR use
    // of `rows` so the 4 KB LDS allocation (and its offset, referenced by the
    // TDM descriptor and the asm ds_loads) cannot be optimized away.
    int opq;
    asm("s_mov_b32 %0, 0" : "=s"(opq));
    if (opq) {
        rows[0][tid] = src[tid];
        rows[1][tid] = src[tid + HH];
    }
#else
    // Fallback staging: coalesced b32 loads through LDS.
    rows[0][tid] = src[(size_t)i   * HH + tid];
    rows[1][tid] = src[(size_t)bot * HH + tid];
#endif
    __syncthreads();

    // Segment boundaries of one ring row (total 4H-4, then 4-wide tail wrap).
    const int b0 = i;
    const int b1 = b0 + el;
    const int b2 = b1 + 2 * i;
    const int b3 = b2 + el;
    const int b4 = b3 + 2 * i;
    const int b5 = b4 + el;
    const int b6 = b5 + 2 * i;
    const int b7 = b6 + el;

    const int jbase = tid << 2;            // 4 outputs per lane, 512 lanes
    v4f o;
#pragma unroll
    for (int k = 0; k < 4; ++k) {
        const int j  = jbase + k;
        const int jj = (j >= WRAP) ? (j - WRAP) : j;   // 4-element tail wrap
        float v;

        const bool gR = (jj >= b2) && (jj < b3);       // right edge (column bot)
        const bool gL = (jj >= b6) && (jj < b7);       // left  edge (column i)
        if (gR || gL) {
            // Column segments: strided global reads; lines are L2-resident
            // across neighboring rings (same cachelines for dif +/- k).
            const int r = gR ? (i + (jj - b2)) : (bot - (jj - b6));
            const int c = gR ? bot : i;
            v = src[(size_t)r * HH + c];
        } else {
            // Row/corner segments: one ds_load from the TDM-staged rows.
            const int bank = ((jj >= b3) && (jj < b6)) ? 1 : 0;
            const int idx  =
                (jj < b0) ? i :                 // (i,i) corner run
                (jj < b1) ? i + (jj - b0) :     // top edge ->
                (jj < b2) ? bot :               // TR corner
                (jj < b4) ? bot :               // BR corner (b3..b4)
                (jj < b5) ? bot - (jj - b4) :   // bottom edge <-
                (jj < b6) ? i :                 // BL corner
                            i;                  // wrap tail (jj >= b7)
            v = lds_read_f32(ldsb + (uint32_t)((bank * HH + idx) << 2));
        }
        o[k] = v;
    }
    // Output is write-once / never re-read: stream past L2 (NT b128 store).
    __builtin_nontemporal_store(o, (v4f*)(dst + jbase));
}

extern "C" void kernel_launch(void* const* d_in, const int* in_sizes, int n_in,
                              void* d_out, int out_size, void* d_ws, size_t ws_size,
                              hipStream_t stream)
{
    (void)n_in; (void)out_size; (void)d_ws; (void)ws_size;
    const float* x   = (const float*)d_in[0];
    float*       out = (float*)d_out;

    const int imgs = in_sizes[0] / (HH * HH);   // B*C = 512
    dim3 grid(imgs * HR);                       // 131072 blocks
    dim3 block(512);                            // 16 wave32s, 4 outputs/lane
    sqrl_gather_kernel<<<grid, block, 0, stream>>>(x, out);
}